// EncoderGraph_25391846654494
// MI455X (gfx1250) — compile-verified
//
#include <hip/hip_runtime.h>
#include <hip/hip_bf16.h>
#include <math.h>

// ---------------------------------------------------------------------------
// GATConv forward on gfx1250 (CDNA5).
//   1) x_lin = X @ W              : f32 WMMA 16x16x4 (full precision)
//   2) a_src/a_dst per node       : wave-per-node dot products
//   3) counting-sort edges by dst : histogram + 1-block scan + id scatter
//   4) warp-per-node aggregation  : softmax (max/denom) + weighted gather,
//      src rows staged in LDS via GLOBAL_LOAD_ASYNC_TO_LDS_B128 (ASYNCcnt),
//      out written once (no f32 atomics at all).
// ---------------------------------------------------------------------------

typedef __attribute__((ext_vector_type(2))) float v2f;
typedef __attribute__((ext_vector_type(8))) float v8f;

constexpr int N_NODES = 100000;
constexpr int N_EDGES = 1600000;
constexpr int E_TOT   = N_EDGES + N_NODES;   // with self loops
constexpr int F       = 256;                 // HEADS*OUT_CH == IN_CH == 256
constexpr int HEADS   = 4;

#if defined(__AMDGCN__) && __has_builtin(__builtin_amdgcn_global_load_async_to_lds_b128) && \
    __has_builtin(__builtin_amdgcn_s_wait_asynccnt)
#define GAT_ASYNC 1
#else
#define GAT_ASYNC 0
#endif

// ------------------------- 1) GEMM via f32 WMMA ----------------------------
// One wave computes two 16x16 output tiles (shares the A fragment).
__global__ void __launch_bounds__(256) gat_gemm_f32(const float* __restrict__ X,
                                                    const float* __restrict__ W,
                                                    float* __restrict__ XL) {
  const int wave = threadIdx.x >> 5;   // 0..7
  const int lane = threadIdx.x & 31;
  const int half = lane >> 4;
  const int r    = lane & 15;
  const int m0   = blockIdx.x * 16;
  const int n0a  = wave * 16;          // first column tile (0..127)
  const int n0b  = n0a + 128;          // second column tile (128..255)

  v8f c0 = {}; v8f c1 = {};
  const float* xrow = X + (size_t)(m0 + r) * F;

  for (int k = 0; k < F; k += 4) {
    const int ka = k + 2 * half;
    v2f a = *(const v2f*)(xrow + ka);
    const float* wr = W + (size_t)ka * F;
    v2f b0, b1;
    b0.x = wr[n0a + r];      b1.x = wr[n0b + r];
    b0.y = wr[F + n0a + r];  b1.y = wr[F + n0b + r];
    c0 = __builtin_amdgcn_wmma_f32_16x16x4_f32(false, a, false, b0, (short)0, c0, false, false);
    c1 = __builtin_amdgcn_wmma_f32_16x16x4_f32(false, a, false, b1, (short)0, c1, false, false);
  }

#pragma unroll
  for (int i = 0; i < 8; ++i) {
    const int row = m0 + i + 8 * half;
    XL[(size_t)row * F + n0a + r] = c0[i];
    XL[(size_t)row * F + n0b + r] = c1[i];
  }
}

// ------------------- 2) per-node attention logits --------------------------
__global__ void __launch_bounds__(256) gat_logits(const float* __restrict__ XL,
                                                  const float* __restrict__ att_src,
                                                  const float* __restrict__ att_dst,
                                                  float* __restrict__ a_src,
                                                  float* __restrict__ a_dst) {
  const int node = (blockIdx.x * blockDim.x + threadIdx.x) >> 5;
  const int lane = threadIdx.x & 31;
  if (node >= N_NODES) return;
  const float* xr = XL + (size_t)node * F + lane * 8;
  const float* as = att_src + lane * 8;
  const float* ad = att_dst + lane * 8;
  float ps = 0.f, pd = 0.f;
#pragma unroll
  for (int i = 0; i < 8; ++i) { const float v = xr[i]; ps += v * as[i]; pd += v * ad[i]; }
  ps += __shfl_xor(ps, 1); pd += __shfl_xor(pd, 1);
  ps += __shfl_xor(ps, 2); pd += __shfl_xor(pd, 2);
  ps += __shfl_xor(ps, 4); pd += __shfl_xor(pd, 4);
  if ((lane & 7) == 0) {
    const int h = lane >> 3;
    a_src[node * HEADS + h] = ps;
    a_dst[node * HEADS + h] = pd;
  }
}

// ------------------- 3) counting sort of edges by dst ----------------------
__device__ __forceinline__ void edge_sd(const long long* __restrict__ ei, int e,
                                        long long& s, long long& d) {
  if (e < N_EDGES) { s = ei[e]; d = ei[N_EDGES + e]; }
  else             { s = d = (long long)(e - N_EDGES); }
}

__global__ void gat_zero_counts(int* __restrict__ counts) {
  const int i = blockIdx.x * 256 + threadIdx.x;
  if (i < N_NODES) counts[i] = 0;
}

__global__ void __launch_bounds__(256) gat_histogram(const long long* __restrict__ ei,
                                                     int* __restrict__ counts) {
  const int e = blockIdx.x * 256 + threadIdx.x;
  if (e >= E_TOT) return;
  long long s, d; edge_sd(ei, e, s, d);
  (void)s;
  atomicAdd(&counts[d], 1);
}

// Single-block chunked Hillis-Steele exclusive scan over N_NODES counts.
__global__ void __launch_bounds__(1024) gat_scan(const int* __restrict__ counts,
                                                 int* __restrict__ row_start,
                                                 int* __restrict__ cursor) {
  __shared__ int buf[1024];
  __shared__ int carry_s;
  const int tid = threadIdx.x;
  if (tid == 0) carry_s = 0;
  __syncthreads();
  for (int base = 0; base < N_NODES; base += 1024) {
    const int i = base + tid;
    const int v = (i < N_NODES) ? counts[i] : 0;
    buf[tid] = v;
    __syncthreads();
    for (int off = 1; off < 1024; off <<= 1) {
      const int t = (tid >= off) ? buf[tid - off] : 0;
      __syncthreads();
      buf[tid] += t;
      __syncthreads();
    }
    const int excl = buf[tid] - v + carry_s;
    if (i < N_NODES) { row_start[i] = excl; cursor[i] = excl; }
    __syncthreads();
    if (tid == 1023) carry_s += buf[1023];
    __syncthreads();
  }
  if (tid == 0) row_start[N_NODES] = E_TOT;
}

__global__ void __launch_bounds__(256) gat_scatter_ids(const long long* __restrict__ ei,
                                                       int* __restrict__ cursor,
                                                       int* __restrict__ edge_ids) {
  const int e = blockIdx.x * 256 + threadIdx.x;
  if (e >= E_TOT) return;
  long long s, d; edge_sd(ei, e, s, d);
  (void)s;
  const int pos = atomicAdd(&cursor[d], 1);
  edge_ids[pos] = e;
}

// ------------------- 4) warp-per-node aggregation --------------------------
#if GAT_ASYNC
// Builtin signature (per hipcc diagnostic): param0 = int4(vector_size 16) AS1*,
// param1 = int4 AS3*, then (int offset, int cpol) immediates.
typedef int v4i_vs __attribute__((vector_size(16)));
typedef v4i_vs __attribute__((address_space(1)))* gptr_v4i;
typedef v4i_vs __attribute__((address_space(3)))* lptr_v4i;

__device__ __forceinline__ void async_row(const float* gsrc, float* lds, int lane) {
  gptr_v4i g = (gptr_v4i)(gsrc + lane * 8);   // 32 bytes per lane
  lptr_v4i l = (lptr_v4i)(lds + lane * 8);
  __builtin_amdgcn_global_load_async_to_lds_b128(g,     l,     0, 0);
  __builtin_amdgcn_global_load_async_to_lds_b128(g + 1, l + 1, 0, 0);
}
#endif

__global__ void __launch_bounds__(256) gat_aggregate(const long long* __restrict__ ei,
                                                     const float* __restrict__ XL,
                                                     const float* __restrict__ a_src,
                                                     const float* __restrict__ a_dst,
                                                     const int* __restrict__ row_start,
                                                     const int* __restrict__ edge_ids,
                                                     const float* __restrict__ bias,
                                                     float* __restrict__ out) {
  const int warp = threadIdx.x >> 5;
  const int lane = threadIdx.x & 31;
  const int n = (blockIdx.x * blockDim.x + threadIdx.x) >> 5;
  if (n >= N_NODES) return;

  const int beg = row_start[n];
  const int end = row_start[n + 1];   // >= beg+1 (self loop guarantees degree >= 1)
  const int h   = lane >> 3;          // head for this lane
  const int ch  = lane * 8;           // first channel for this lane
  const float adst = a_dst[n * HEADS + h];

  // ---- pass 1a: per-head segment max (8 lanes per head, stride-8 over edges)
  float mx = -INFINITY;
  for (int j = beg + (lane & 7); j < end; j += 8) {
    const int eid = edge_ids[j];
    const long long s = (eid < N_EDGES) ? ei[eid] : (long long)n;
    float al = a_src[s * HEADS + h] + adst;
    al = al > 0.f ? al : 0.2f * al;
    mx = fmaxf(mx, al);
  }
  mx = fmaxf(mx, __shfl_xor(mx, 1));
  mx = fmaxf(mx, __shfl_xor(mx, 2));
  mx = fmaxf(mx, __shfl_xor(mx, 4));

  // ---- pass 1b: per-head denominator
  float dsum = 0.f;
  for (int j = beg + (lane & 7); j < end; j += 8) {
    const int eid = edge_ids[j];
    const long long s = (eid < N_EDGES) ? ei[eid] : (long long)n;
    float al = a_src[s * HEADS + h] + adst;
    al = al > 0.f ? al : 0.2f * al;
    dsum += __expf(al - mx);
  }
  dsum += __shfl_xor(dsum, 1);
  dsum += __shfl_xor(dsum, 2);
  dsum += __shfl_xor(dsum, 4);
  const float inv = 1.f / (dsum + 1e-16f);

  // ---- pass 2: weighted accumulation of source rows (registers only)
  float acc[8] = {0.f, 0.f, 0.f, 0.f, 0.f, 0.f, 0.f, 0.f};

#if GAT_ASYNC
  __shared__ float rows[8][2][F];     // 16 KB: double buffer per warp
  float* buf0 = &rows[warp][0][0];
  float* buf1 = &rows[warp][1][0];

  {
    const int eid0 = edge_ids[beg];
    long long s_cur = (eid0 < N_EDGES) ? ei[eid0] : (long long)n;
    async_row(XL + (size_t)s_cur * F, buf0, lane);
    int cur = 0;
    for (int j = beg; j < end; ++j) {
      long long s_nxt = 0;
      if (j + 1 < end) {
        const int eid1 = edge_ids[j + 1];
        s_nxt = (eid1 < N_EDGES) ? ei[eid1] : (long long)n;
        async_row(XL + (size_t)s_nxt * F, cur ? buf0 : buf1, lane);
        __builtin_amdgcn_s_wait_asynccnt(2);   // only newest prefetch in flight
      } else {
        __builtin_amdgcn_s_wait_asynccnt(0);
      }
      float al = a_src[s_cur * HEADS + h] + adst;
      al = al > 0.f ? al : 0.2f * al;
      const float w = __expf(al - mx) * inv;
      const float* row = (cur ? buf1 : buf0) + ch;
#pragma unroll
      for (int i = 0; i < 8; ++i) acc[i] += w * row[i];
      s_cur = s_nxt;
      cur ^= 1;
    }
  }
#else
  for (int j = beg; j < end; ++j) {
    const int eid = edge_ids[j];
    const long long s = (eid < N_EDGES) ? ei[eid] : (long long)n;
    float al = a_src[s * HEADS + h] + adst;
    al = al > 0.f ? al : 0.2f * al;
    const float w = __expf(al - mx) * inv;
    const float4* x4 = (const float4*)(XL + (size_t)s * F + ch);
    const float4 v0 = x4[0], v1 = x4[1];
    acc[0] += w * v0.x; acc[1] += w * v0.y; acc[2] += w * v0.z; acc[3] += w * v0.w;
    acc[4] += w * v1.x; acc[5] += w * v1.y; acc[6] += w * v1.z; acc[7] += w * v1.w;
  }
#endif

  const float* bp = bias + ch;
  float* op = out + (size_t)n * F + ch;
#pragma unroll
  for (int i = 0; i < 8; ++i) op[i] = acc[i] + bp[i];
}

// ---------------------------------------------------------------------------
extern "C" void kernel_launch(void* const* d_in, const int* in_sizes, int n_in,
                              void* d_out, int out_size, void* d_ws, size_t ws_size,
                              hipStream_t stream) {
  (void)in_sizes; (void)n_in; (void)out_size; (void)ws_size;
  const float*     x        = (const float*)d_in[0];
  const long long* ei       = (const long long*)d_in[1];
  const float*     Wm       = (const float*)d_in[2];
  const float*     att_src  = (const float*)d_in[3];
  const float*     att_dst  = (const float*)d_in[4];
  const float*     bias     = (const float*)d_in[5];
  float*           out      = (float*)d_out;

  // workspace layout
  float* XL       = (float*)d_ws;                          // N*256 f32
  float* a_src    = XL + (size_t)N_NODES * F;              // N*4
  float* a_dst    = a_src + (size_t)N_NODES * HEADS;       // N*4
  int*   counts   = (int*)(a_dst + (size_t)N_NODES * HEADS); // N
  int*   row_start= counts + N_NODES;                      // N+1
  int*   cursor   = row_start + (N_NODES + 1);             // N
  int*   edge_ids = cursor + N_NODES;                      // E_TOT

  // 1) GEMM (WMMA): 6250 row tiles, 8 waves/block cover all 16 column tiles
  gat_gemm_f32<<<N_NODES / 16, 256, 0, stream>>>(x, Wm, XL);

  // 2) logits: wave per node
  gat_logits<<<(N_NODES * 32 + 255) / 256, 256, 0, stream>>>(XL, att_src, att_dst, a_src, a_dst);

  // 3) counting sort by destination
  gat_zero_counts<<<(N_NODES + 255) / 256, 256, 0, stream>>>(counts);
  gat_histogram<<<(E_TOT + 255) / 256, 256, 0, stream>>>(ei, counts);
  gat_scan<<<1, 1024, 0, stream>>>(counts, row_start, cursor);
  gat_scatter_ids<<<(E_TOT + 255) / 256, 256, 0, stream>>>(ei, cursor, edge_ids);

  // 4) warp-per-node softmax + aggregation, out written once
  gat_aggregate<<<(N_NODES * 32 + 255) / 256, 256, 0, stream>>>(
      ei, XL, a_src, a_dst, row_start, edge_ids, bias, out);
}